// Model_34162169872715
// MI455X (gfx1250) — compile-verified
//
#include <hip/hip_runtime.h>
#include <hip/hip_bf16.h>

#define SEQ_LEN 65536
#define STEPS   (SEQ_LEN - 1)     // 65535 recurrence steps
#define H 30
#define P 25
#define ROWS SEQ_LEN              // Hbuf rows (row 65535 zero-padded)
#define TILES (ROWS / 16)         // 4096 WMMA row-tiles
#define TILES_PER_WAVE 4

typedef float v2f __attribute__((ext_vector_type(2)));
typedef float v8f __attribute__((ext_vector_type(8)));

__device__ __forceinline__ float lanebcast(float v, int lane) {
  return __builtin_bit_cast(float,
      __builtin_amdgcn_readlane(__builtin_bit_cast(unsigned int, v),
                                (unsigned int)lane));
}

__device__ __forceinline__ float fast_sigmoid(float x) {
  // 1/(1+exp(-x)) via v_exp_f32 (exp2) + v_rcp_f32
  float e = __builtin_amdgcn_exp2f(x * -1.44269504088896f);
  return __builtin_amdgcn_rcpf(1.0f + e);
}

__device__ __forceinline__ float fast_tanh(float x) {
  // 1 - 2/(exp(2x)+1); saturates correctly via exp2 overflow/underflow
  float e = __builtin_amdgcn_exp2f(x * 2.88539008177793f);
  return 1.0f - 2.0f * __builtin_amdgcn_rcpf(1.0f + e);
}

__device__ __forceinline__ v8f wmma4(v2f a, v2f b, v8f c) {
  return __builtin_amdgcn_wmma_f32_16x16x4_f32(false, a, false, b,
                                               (short)0, c, false, false);
}

// ---------------------------------------------------------------------------
// Phase 0: zero-pad head weights into WMMA-friendly buffers (one wave, once).
// W1p[n*32+k] = W1^T padded to 32x32; b1p/W2p padded to 32.
// ---------------------------------------------------------------------------
__global__ __launch_bounds__(32) void head_prep(
    const float* __restrict__ W1, const float* __restrict__ b1,
    const float* __restrict__ W2,
    float* __restrict__ W1p, float* __restrict__ b1p, float* __restrict__ W2p)
{
  const int j = threadIdx.x;                    // n (output col) 0..31
  #pragma unroll
  for (int k = 0; k < 32; ++k)
    W1p[j * 32 + k] = (j < H && k < H) ? W1[j * H + k] : 0.0f;
  b1p[j] = (j < H) ? b1[j] : 0.0f;
  W2p[j] = (j < H) ? W2[j] : 0.0f;
}

// ---------------------------------------------------------------------------
// Phase 1: the serial GRU recurrence. One wave32. Lane j owns gate rows
// j, 30+j, 60+j of W_hh in VGPRs; h[j] in a register; h broadcast via
// v_readlane. Stores h_t (padded to 32 cols, cols 30/31 = 0) to Hbuf.
// ---------------------------------------------------------------------------
__global__ __launch_bounds__(32) void gru_recurrence(
    const float* __restrict__ params, const float* __restrict__ seq,
    const float* __restrict__ phi_W,  const float* __restrict__ phi_b,
    const float* __restrict__ W_ih,   const float* __restrict__ b_ih,
    const float* __restrict__ W_hh,   const float* __restrict__ b_hh,
    float* __restrict__ Hbuf)
{
  const int j   = threadIdx.x;
  const int row = (j < H) ? j : (H - 1);   // clamp lanes 30/31 for loads

  // phi = relu(params @ phi_W.T + phi_b)   (lane-uniform, one-time)
  float phi[P];
  #pragma unroll
  for (int p = 0; p < P; ++p) {
    float a = phi_b[p];
    #pragma unroll
    for (int q = 0; q < 11; ++q) a = fmaf(params[q], phi_W[p * 11 + q], a);
    phi[p] = a > 0.0f ? a : 0.0f;
  }

  // Fold phi part of gi (and both biases for r,z) into per-lane constants.
  float c_r = b_ih[row]         + b_hh[row];
  float c_z = b_ih[H + row]     + b_hh[H + row];
  float c_n = b_ih[2 * H + row];           // b_hh_n must stay separate (r*gh_n)
  float bn  = b_hh[2 * H + row];
  #pragma unroll
  for (int p = 0; p < P; ++p) {
    c_r = fmaf(W_ih[row * (P + 1) + p],           phi[p], c_r);
    c_z = fmaf(W_ih[(H + row) * (P + 1) + p],     phi[p], c_z);
    c_n = fmaf(W_ih[(2 * H + row) * (P + 1) + p], phi[p], c_n);
  }
  const float wl_r = W_ih[row * (P + 1) + P];
  const float wl_z = W_ih[(H + row) * (P + 1) + P];
  const float wl_n = W_ih[(2 * H + row) * (P + 1) + P];

  // Resident W_hh rows: 90 VGPRs.
  float Wr[H], Wz[H], Wn[H];
  #pragma unroll
  for (int k = 0; k < H; ++k) {
    Wr[k] = W_hh[row * H + k];
    Wz[k] = W_hh[(H + row) * H + k];
    Wn[k] = W_hh[(2 * H + row) * H + k];
  }

  float h = 0.0f;

  #pragma unroll 1
  for (int t0 = 0; t0 < STEPS; t0 += 32) {
    float sv = seq[t0 + j];                     // t0+j <= 65535: in range
    if (t0 + 32 < STEPS) __builtin_prefetch(&seq[t0 + 32 + j], 0, 0);
    const int inner = (STEPS - t0) < 32 ? (STEPS - t0) : 32;
    #pragma unroll 1
    for (int i = 0; i < inner; ++i) {
      const float s_prev = lanebcast(sv, i);
      float dr0 = fmaf(wl_r, s_prev, c_r), dr1 = 0.0f;
      float dz0 = fmaf(wl_z, s_prev, c_z), dz1 = 0.0f;
      float dn0 = bn,                      dn1 = 0.0f;
      #pragma unroll
      for (int k = 0; k < H; k += 2) {
        const float h0 = lanebcast(h, k);
        const float h1 = lanebcast(h, k + 1);
        dr0 = fmaf(Wr[k],     h0, dr0);
        dz0 = fmaf(Wz[k],     h0, dz0);
        dn0 = fmaf(Wn[k],     h0, dn0);
        dr1 = fmaf(Wr[k + 1], h1, dr1);
        dz1 = fmaf(Wz[k + 1], h1, dz1);
        dn1 = fmaf(Wn[k + 1], h1, dn1);
      }
      const float r  = fast_sigmoid(dr0 + dr1);
      const float z  = fast_sigmoid(dz0 + dz1);
      const float gn = fmaf(wl_n, s_prev, c_n);
      const float n  = fast_tanh(fmaf(r, dn0 + dn1, gn));
      const float hn = fmaf(z, h - n, n);
      h = (j < H) ? hn : 0.0f;                  // keep pad cols 30/31 zero
      Hbuf[(size_t)(t0 + i) * 32 + j] = h;      // 128B coalesced store
    }
  }
  Hbuf[(size_t)(ROWS - 1) * 32 + j] = 0.0f;     // zero-pad last tile row
}

// ---------------------------------------------------------------------------
// Phase 2: head MLP over all timesteps as a WMMA GEMM. One wave handles
// TILES_PER_WAVE 16-row tiles with B fragments (padded W1^T) resident in
// VGPRs. Branch-free epilogue: relu + W2 contraction + shfl_xor reduction.
// ---------------------------------------------------------------------------
__global__ __launch_bounds__(32) void head_wmma(
    const float* __restrict__ Hbuf, const float* __restrict__ seq,
    const float* __restrict__ W1p,  const float* __restrict__ b1p,
    const float* __restrict__ W2p,  const float* __restrict__ b2,
    float* __restrict__ part)
{
  const int L  = threadIdx.x;
  const int nl = L & 15;                        // M for A-frag, N for B/C-frag
  const int hi = L >> 4;                        // half-wave select
  const int n1 = nl + 16;                       // second N tile

  // Resident B fragments (padded -> unconditional contiguous b64 loads).
  v2f B0[8], B1[8];
  #pragma unroll
  for (int kc = 0; kc < 8; ++kc) {
    const int k0 = kc * 4 + 2 * hi;
    B0[kc].x = W1p[nl * 32 + k0];
    B0[kc].y = W1p[nl * 32 + k0 + 1];
    B1[kc].x = W1p[n1 * 32 + k0];
    B1[kc].y = W1p[n1 * 32 + k0 + 1];
  }
  const float b1n0  = b1p[nl];
  const float w2n0  = W2p[nl];
  const float b1n1  = b1p[n1];
  const float w2n1  = W2p[n1];
  const float bias2 = b2[0];

  const int tile0 = blockIdx.x * TILES_PER_WAVE;
  #pragma unroll 1
  for (int tt = 0; tt < TILES_PER_WAVE; ++tt) {
    const int tile = tile0 + tt;
    const float* arow = Hbuf + (size_t)(tile * 16 + nl) * 32 + 2 * hi;

    v8f c0 = {}; v8f c1 = {};
    #pragma unroll
    for (int kc = 0; kc < 8; ++kc) {
      v2f a;
      a.x = arow[kc * 4];                       // pad cols 30/31 already zero
      a.y = arow[kc * 4 + 1];
      c0 = wmma4(a, B0[kc], c0);
      c1 = wmma4(a, B1[kc], c1);
    }

    float loss = 0.0f;
    #pragma unroll
    for (int v = 0; v < 8; ++v) {
      float q = fmaxf(c0[v] + b1n0, 0.0f) * w2n0 +
                fmaxf(c1[v] + b1n1, 0.0f) * w2n1;
      q += __shfl_xor(q, 1, 32);
      q += __shfl_xor(q, 2, 32);
      q += __shfl_xor(q, 4, 32);
      q += __shfl_xor(q, 8, 32);                // q uniform per 16-lane half
      const int r     = tile * 16 + v + 8 * hi; // C layout: M = v + 8*half
      const int valid = (r < STEPS);
      const int ti    = valid ? (r + 1) : 0;    // clamp address, mask value
      const float d   = (q + bias2) - seq[ti];
      loss += valid ? d * d : 0.0f;             // v_cndmask, no branch
    }
    // loss identical across each 16-lane half: combine halves, lane0 writes.
    const float lo = lanebcast(loss, 0);
    const float hi16 = lanebcast(loss, 16);
    if (L == 0) part[tile] = lo + hi16;
  }
}

// ---------------------------------------------------------------------------
// Phase 3: deterministic fixed-order reduction of 4096 partials -> out[0].
// ---------------------------------------------------------------------------
__global__ __launch_bounds__(256) void reduce_partials(
    const float* __restrict__ part, float* __restrict__ out)
{
  __shared__ float sm[256];
  const int t = threadIdx.x;
  float a = 0.0f;
  for (int i = t; i < TILES; i += 256) a += part[i];
  sm[t] = a;
  __syncthreads();
  for (int s = 128; s > 0; s >>= 1) {
    if (t < s) sm[t] += sm[t + s];
    __syncthreads();
  }
  if (t == 0) out[0] = sm[0] * (1.0f / (float)SEQ_LEN);
}

extern "C" void kernel_launch(void* const* d_in, const int* in_sizes, int n_in,
                              void* d_out, int out_size, void* d_ws, size_t ws_size,
                              hipStream_t stream) {
  (void)in_sizes; (void)n_in; (void)out_size; (void)ws_size;
  const float* params  = (const float*)d_in[0];
  const float* seq     = (const float*)d_in[1];
  const float* phi_W   = (const float*)d_in[2];
  const float* phi_b   = (const float*)d_in[3];
  const float* W_ih    = (const float*)d_in[4];
  const float* b_ih    = (const float*)d_in[5];
  const float* W_hh    = (const float*)d_in[6];
  const float* b_hh    = (const float*)d_in[7];
  const float* head_W1 = (const float*)d_in[8];
  const float* head_b1 = (const float*)d_in[9];
  const float* head_W2 = (const float*)d_in[10];
  const float* head_b2 = (const float*)d_in[11];
  float* out = (float*)d_out;

  float* Hbuf = (float*)d_ws;                    // ROWS*32 floats = 8 MB
  float* part = Hbuf + (size_t)ROWS * 32;        // TILES floats
  float* W1p  = part + TILES;                    // 32*32 floats
  float* b1p  = W1p + 32 * 32;                   // 32 floats
  float* W2p  = b1p + 32;                        // 32 floats

  head_prep<<<1, 32, 0, stream>>>(head_W1, head_b1, head_W2, W1p, b1p, W2p);
  gru_recurrence<<<1, 32, 0, stream>>>(params, seq, phi_W, phi_b,
                                       W_ih, b_ih, W_hh, b_hh, Hbuf);
  head_wmma<<<TILES / TILES_PER_WAVE, 32, 0, stream>>>(
      Hbuf, seq, W1p, b1p, W2p, head_b2, part);
  reduce_partials<<<1, 256, 0, stream>>>(part, out);
}